// Low_Fidelity_FusionDeepONet_11836929868249
// MI455X (gfx1250) — compile-verified
//
#include <hip/hip_runtime.h>

typedef float v2f __attribute__((ext_vector_type(2)));
typedef float v8f __attribute__((ext_vector_type(8)));

#define BATCH   16
#define NPTS    10000
#define HID     128
#define NLAYERS 4
#define NOUT    5
#define TRS     132                 // padded LDS row stride (floats): 132 % 64 = 4 -> conflict-free A loads
#define WAVES   4
#define RPW     16                  // rows per wave (WMMA M)
#define TILE_ROWS (WAVES * RPW)     // 64 points per block

// ---------------------------------------------------------------------------
// Fast activation: CDNA5 has native V_TANH_F32 / V_SIN_F32 in the TRANS pipe
// (co-executes with WMMA). Avoid the branchy libm lowering at all costs.
// ---------------------------------------------------------------------------
__device__ __forceinline__ float fast_tanh(float x) {
#if __has_builtin(__builtin_amdgcn_tanhf)
    return __builtin_amdgcn_tanhf(x);
#else
    const float e = __expf(2.0f * fabsf(x));                 // v_exp_f32 path
    const float r = 1.0f - 2.0f * __builtin_amdgcn_rcpf(e + 1.0f);
    return copysignf(r, x);
#endif
}

__device__ __forceinline__ float rowdy(float t, float a, float w) {
    return fast_tanh(t) + a * __sinf(w * t);                 // v_sin_f32, args are O(1)
}

// ---------------------------------------------------------------------------
// Phase 1: branch MLP + fusion gates + branch_out. Tiny (16 rows) -> 1 block.
// gates layout: [layer 0..3][b 0..15][hid 0..127]; bc layout: [b][o*128+h]
// ---------------------------------------------------------------------------
__global__ __launch_bounds__(HID)
void branch_kernel(const float* __restrict__ params,
                   const float* __restrict__ Wb0, const float* __restrict__ bb0,
                   const float* __restrict__ Wb,  const float* __restrict__ bb,
                   const float* __restrict__ Wbf, const float* __restrict__ bbf,
                   const float* __restrict__ ab,  const float* __restrict__ wb,
                   float* __restrict__ gates, float* __restrict__ bc)
{
    __shared__ float hbuf[BATCH][HID];
    __shared__ float sp[BATCH][6];
    const int j = threadIdx.x;                 // hidden unit owned by this thread
    if (j < BATCH * 6) sp[j / 6][j % 6] = params[j];
    __syncthreads();

    float hcur[BATCH], gsum[BATCH];
    // layer 0: params(16x6) @ Wb0(6x128)
    for (int b = 0; b < BATCH; ++b) {
        float acc = bb0[j];
        #pragma unroll
        for (int k = 0; k < 6; ++k) acc += sp[b][k] * Wb0[k * HID + j];
        float z = rowdy(acc, ab[j], wb[j]);
        hcur[b] = z; gsum[b] = z;
        gates[(0 * BATCH + b) * HID + j] = z;
        hbuf[b][j] = z;
    }
    __syncthreads();

    for (int l = 0; l < NLAYERS - 1; ++l) {
        const float* W = Wb + (size_t)l * HID * HID;
        for (int b = 0; b < BATCH; ++b) {
            float acc = bb[l * HID + j];
            for (int k = 0; k < HID; ++k) acc += hbuf[b][k] * W[k * HID + j];
            float z = rowdy(acc, ab[(l + 1) * HID + j], wb[(l + 1) * HID + j]);
            hcur[b] = z; gsum[b] += z;
            gates[((l + 1) * BATCH + b) * HID + j] = gsum[b];
        }
        __syncthreads();                       // everyone done reading hbuf
        for (int b = 0; b < BATCH; ++b) hbuf[b][j] = hcur[b];
        __syncthreads();
    }

    // branch_out = h3 @ Wbf(128x640) + bbf ; thread j owns cols j, j+128, ...
    for (int o = 0; o < NOUT; ++o) {
        const int c = o * HID + j;
        for (int b = 0; b < BATCH; ++b) {
            float acc = bbf[c];
            for (int k = 0; k < HID; ++k) acc += hbuf[b][k] * Wbf[k * (HID * NOUT) + c];
            bc[b * (HID * NOUT) + c] = acc;
        }
    }
}

// ---------------------------------------------------------------------------
// Phase 2: trunk MLP (fp32 WMMA 16x16x4) + gate fusion + WMMA contraction.
// Each wave owns 16 points; block = 4 waves = 64 points; grid = (157, 16).
// A-fragment layout (ISA 32-bit A 16x4): lanes 0-15 -> K=0,1 ; lanes 16-31 -> K=2,3.
// B-fragment layout (mirror):            lanes 0-15 -> K=0,1 at N=lane ; 16-31 -> K=2,3.
// C/D layout: reg r, lanes 0-15 -> (M=r, N=lane); lanes 16-31 -> (M=r+8, N=lane-16).
// ---------------------------------------------------------------------------
__global__ __launch_bounds__(WAVES * 32)
void trunk_kernel(const float* __restrict__ coords, const float* __restrict__ sdf,
                  const float* __restrict__ Wt0, const float* __restrict__ bt0,
                  const float* __restrict__ Wt,  const float* __restrict__ bt,
                  const float* __restrict__ Wtf, const float* __restrict__ btf,
                  const float* __restrict__ at,  const float* __restrict__ wt,
                  const float* __restrict__ gates, const float* __restrict__ bc,
                  float* __restrict__ out)
{
    __shared__ float xs[WAVES * RPW * TRS];
    const int lane = threadIdx.x & 31;
    const int wave = threadIdx.x >> 5;
    const int b    = blockIdx.y;
    const int row0 = blockIdx.x * TILE_ROWS + wave * RPW;
    float* xw = xs + wave * RPW * TRS;

    const int rowA  = lane & 15;               // matrix row this lane feeds (A)
    const int koff  = (lane >> 4) << 1;        // 0 for lanes 0-15, 2 for 16-31
    const int colC  = lane & 15;               // column within N-tile (C/D, B)
    const int rhalf = (lane >> 4) << 3;        // +8 row offset for upper half-wave

    // ---- layer 0: K = 4, input row = [cx, cy, cz, sdf] ----
    int pt = row0 + rowA; if (pt > NPTS - 1) pt = NPTS - 1;   // clamp tail (stores guarded later)
    const float* cp = coords + ((size_t)b * NPTS + pt) * 3;
    const float sv  = sdf[(size_t)b * NPTS + pt];
    v2f afrag;
    afrag.x = (lane < 16) ? cp[0] : cp[2];
    afrag.y = (lane < 16) ? cp[1] : sv;

    {
        const float* g = gates + (0 * BATCH + b) * HID;
        #pragma unroll
        for (int nt = 0; nt < 8; ++nt) {
            const int col = nt * 16 + colC;
            v2f bfrag;
            bfrag.x = Wt0[(koff    ) * HID + col];
            bfrag.y = Wt0[(koff + 1) * HID + col];
            v8f acc = {};
            acc = __builtin_amdgcn_wmma_f32_16x16x4_f32(false, afrag, false, bfrag,
                                                        (short)0, acc, false, false);
            const float bias = bt0[col], av = at[col], wv = wt[col], gv = g[col];
            #pragma unroll
            for (int r = 0; r < 8; ++r) {
                const float t = acc[r] + bias;
                xw[(r + rhalf) * TRS + col] = rowdy(t, av, wv) * gv;
            }
        }
    }
    __syncthreads();

    // ---- hidden layers 1..3: 128x128 GEMM per wave-tile ----
    for (int l = 0; l < NLAYERS - 1; ++l) {
        const float* W = Wt + (size_t)l * HID * HID;
        v8f zero = {};
        v8f accs[8];
        #pragma unroll
        for (int nt = 0; nt < 8; ++nt) accs[nt] = zero;

        for (int kk = 0; kk < HID; kk += 4) {
            const v2f a = *(const v2f*)(xw + rowA * TRS + kk + koff);  // reused by all 8 N-tiles
            #pragma unroll
            for (int nt = 0; nt < 8; ++nt) {
                const int col = nt * 16 + colC;
                v2f bfrag;
                bfrag.x = W[(kk + koff    ) * HID + col];
                bfrag.y = W[(kk + koff + 1) * HID + col];
                accs[nt] = __builtin_amdgcn_wmma_f32_16x16x4_f32(false, a, false, bfrag,
                                                                 (short)0, accs[nt], false, false);
            }
        }
        __syncthreads();                        // all K-reads of xw complete
        const float* g   = gates + ((l + 1) * BATCH + b) * HID;
        const float* aa  = at + (l + 1) * HID;
        const float* ww  = wt + (l + 1) * HID;
        const float* bbp = bt + l * HID;
        #pragma unroll
        for (int nt = 0; nt < 8; ++nt) {
            const int col = nt * 16 + colC;
            const float bias = bbp[col], av = aa[col], wv = ww[col], gv = g[col];
            #pragma unroll
            for (int r = 0; r < 8; ++r) {
                const float t = accs[nt][r] + bias;
                xw[(r + rhalf) * TRS + col] = rowdy(t, av, wv) * gv;
            }
        }
        __syncthreads();
    }

    // ---- final linear: Wtf + btf (no activation, no gate) ----
    {
        v8f zero = {};
        v8f accs[8];
        #pragma unroll
        for (int nt = 0; nt < 8; ++nt) accs[nt] = zero;
        for (int kk = 0; kk < HID; kk += 4) {
            const v2f a = *(const v2f*)(xw + rowA * TRS + kk + koff);
            #pragma unroll
            for (int nt = 0; nt < 8; ++nt) {
                const int col = nt * 16 + colC;
                v2f bfrag;
                bfrag.x = Wtf[(kk + koff    ) * HID + col];
                bfrag.y = Wtf[(kk + koff + 1) * HID + col];
                accs[nt] = __builtin_amdgcn_wmma_f32_16x16x4_f32(false, a, false, bfrag,
                                                                 (short)0, accs[nt], false, false);
            }
        }
        __syncthreads();
        #pragma unroll
        for (int nt = 0; nt < 8; ++nt) {
            const int col = nt * 16 + colC;
            const float bias = btf[col];
            #pragma unroll
            for (int r = 0; r < 8; ++r)
                xw[(r + rhalf) * TRS + col] = accs[nt][r] + bias;
        }
        __syncthreads();
    }

    // ---- final contraction on WMMA: trunk(16x128) @ bc^T(128x16), cols 0..4 valid ----
    {
        const float* bcb = bc + (size_t)b * HID * NOUT;
        const int oc = (colC < NOUT) ? colC : 0;      // bc column this lane supplies (dupes unused)
        v8f acc = {};
        for (int kk = 0; kk < HID; kk += 4) {
            const v2f a     = *(const v2f*)(xw + rowA * TRS + kk + koff);
            const v2f bfrag = *(const v2f*)(bcb + oc * HID + kk + koff);
            acc = __builtin_amdgcn_wmma_f32_16x16x4_f32(false, a, false, bfrag,
                                                        (short)0, acc, false, false);
        }
        // predicated stores AFTER all matrix ops (EXEC all-ones during WMMA)
        if (colC < NOUT) {
            #pragma unroll
            for (int r = 0; r < 8; ++r) {
                const int p = row0 + r + rhalf;
                if (p < NPTS)
                    out[((size_t)b * NPTS + p) * NOUT + colC] = acc[r];
            }
        }
    }
}

// ---------------------------------------------------------------------------
extern "C" void kernel_launch(void* const* d_in, const int* in_sizes, int n_in,
                              void* d_out, int out_size, void* d_ws, size_t ws_size,
                              hipStream_t stream) {
    const float* coords = (const float*)d_in[0];
    const float* sdf    = (const float*)d_in[1];
    const float* params = (const float*)d_in[2];
    const float* Wb0    = (const float*)d_in[3];
    const float* bb0    = (const float*)d_in[4];
    const float* Wb     = (const float*)d_in[5];
    const float* bb     = (const float*)d_in[6];
    const float* Wbf    = (const float*)d_in[7];
    const float* bbf    = (const float*)d_in[8];
    const float* ab     = (const float*)d_in[9];
    const float* wb     = (const float*)d_in[10];
    const float* Wt0    = (const float*)d_in[11];
    const float* bt0    = (const float*)d_in[12];
    const float* Wt     = (const float*)d_in[13];
    const float* bt     = (const float*)d_in[14];
    const float* Wtf    = (const float*)d_in[15];
    const float* btf    = (const float*)d_in[16];
    const float* at     = (const float*)d_in[17];
    const float* wt     = (const float*)d_in[18];

    float* ws    = (float*)d_ws;
    float* gates = ws;                               // 4*16*128 floats
    float* bc    = ws + NLAYERS * BATCH * HID;       // 16*640 floats

    branch_kernel<<<1, HID, 0, stream>>>(params, Wb0, bb0, Wb, bb, Wbf, bbf, ab, wb,
                                         gates, bc);

    dim3 grid((NPTS + TILE_ROWS - 1) / TILE_ROWS, BATCH);
    trunk_kernel<<<grid, WAVES * 32, 0, stream>>>(coords, sdf, Wt0, bt0, Wt, bt,
                                                  Wtf, btf, at, wt, gates, bc,
                                                  (float*)d_out);
}